// TensorInteraction_9216999817578
// MI455X (gfx1250) — compile-verified
//
#include <hip/hip_runtime.h>
#include <hip/hip_bf16.h>

// ---------------------------------------------------------------------------
// PaiNN-style tensor interaction for gfx1250 (MI455X).
// Dense stages use V_WMMA_F32_16X16X4_F32 (fp32 in/out). Each wave computes a
// 16x64 output tile (4 WMMA accumulators sharing one A fragment). Edge
// gather/scatter stage is bandwidth-bound and uses global_atomic_add_f32.
// ---------------------------------------------------------------------------

typedef __attribute__((ext_vector_type(2))) float v2f;
typedef __attribute__((ext_vector_type(8))) float v8f;

#define EPI_NONE   0
#define EPI_SWISH  1
#define EPI_RADIAL 2

// D[row,col] = epi( sum_k A[row,k]*B[k,col] + bias[col] )
//   A addressed as A[a_base + row*a_rs + k*a_cs]
//   B dense row-major [K x Ncols]; Ncols must be a multiple of 64
//   D addressed as D[d_base + row*d_rs + col*d_cs]
// One wave -> 16 rows x 64 cols (4 x 16x16 WMMA tiles, K stepped by 4).
template <int EPI, bool ACONTIG>
__global__ void wmma_gemm_f32(const float* __restrict__ A, long long a_base,
                              int a_rs, int a_cs,
                              const float* __restrict__ B,
                              const float* __restrict__ bias,
                              const float* __restrict__ rscale,
                              float* __restrict__ D, long long d_base,
                              int d_rs, int d_cs,
                              int M, int K, int Ncols) {
  const int wave = threadIdx.x >> 5;
  const int lane = threadIdx.x & 31;
  const int ct   = Ncols >> 6;                       // 64-wide column groups
  const int tile = blockIdx.x * (blockDim.x >> 5) + wave;
  const int ntiles = (M >> 4) * ct;
  if (tile >= ntiles) return;                        // wave-uniform: EXEC stays all-1s

  const int row0 = (tile / ct) << 4;
  const int col0 = (tile % ct) << 6;
  const int hl   = lane >> 4;                        // which K-half this lane holds
  const int lidx = lane & 15;
  const int m = row0 + lidx;                         // A row for this lane
  const int n = col0 + lidx;                         // B col (tile 0) for this lane

  v8f acc0 = {0.f, 0.f, 0.f, 0.f, 0.f, 0.f, 0.f, 0.f};
  v8f acc1 = acc0, acc2 = acc0, acc3 = acc0;

  // Lane-private streaming pointers: A at K = 2*hl, B row K = 2*hl.
  const float* pA = A + a_base + (long long)m * a_rs + (long long)(2 * hl) * a_cs;
  const float* pB = B + (long long)(2 * hl) * Ncols + n;

  const int Kmain = K & ~3;
#pragma unroll 2
  for (int k = 0; k < Kmain; k += 4) {
    v2f a;
    if (ACONTIG) {
      a = *(const v2f*)pA;                           // b64 load (8B aligned)
    } else {
      a[0] = pA[0];
      a[1] = pA[a_cs];
    }
    v2f b0, b1, b2, b3;
    b0[0] = pB[0];          b0[1] = pB[Ncols];
    b1[0] = pB[16];         b1[1] = pB[Ncols + 16];
    b2[0] = pB[32];         b2[1] = pB[Ncols + 32];
    b3[0] = pB[48];         b3[1] = pB[Ncols + 48];
    acc0 = __builtin_amdgcn_wmma_f32_16x16x4_f32(false, a, false, b0, (short)0, acc0, false, false);
    acc1 = __builtin_amdgcn_wmma_f32_16x16x4_f32(false, a, false, b1, (short)0, acc1, false, false);
    acc2 = __builtin_amdgcn_wmma_f32_16x16x4_f32(false, a, false, b2, (short)0, acc2, false, false);
    acc3 = __builtin_amdgcn_wmma_f32_16x16x4_f32(false, a, false, b3, (short)0, acc3, false, false);
    pA += ACONTIG ? 4 : 4 * a_cs;
    pB += 4 * Ncols;
  }

  if (Kmain < K) {                                   // guarded tail (K % 4 != 0)
    const int ka = Kmain + 2 * hl;
    v2f a = {0.f, 0.f};
    v2f b0 = {0.f, 0.f}, b1 = b0, b2 = b0, b3 = b0;
    if (ka < K) {
      a[0] = pA[0];
      b0[0] = pB[0]; b1[0] = pB[16]; b2[0] = pB[32]; b3[0] = pB[48];
    }
    if (ka + 1 < K) {
      a[1] = ACONTIG ? pA[1] : pA[a_cs];
      b0[1] = pB[Ncols]; b1[1] = pB[Ncols + 16];
      b2[1] = pB[Ncols + 32]; b3[1] = pB[Ncols + 48];
    }
    acc0 = __builtin_amdgcn_wmma_f32_16x16x4_f32(false, a, false, b0, (short)0, acc0, false, false);
    acc1 = __builtin_amdgcn_wmma_f32_16x16x4_f32(false, a, false, b1, (short)0, acc1, false, false);
    acc2 = __builtin_amdgcn_wmma_f32_16x16x4_f32(false, a, false, b2, (short)0, acc2, false, false);
    acc3 = __builtin_amdgcn_wmma_f32_16x16x4_f32(false, a, false, b3, (short)0, acc3, false, false);
  }

  // C/D layout: VGPR r -> row r (lanes 0-15) / row r+8 (lanes 16-31), col = lane&15
  v8f accs[4] = {acc0, acc1, acc2, acc3};
#pragma unroll
  for (int t = 0; t < 4; ++t) {
#pragma unroll
    for (int r = 0; r < 8; ++r) {
      const int row = row0 + r + 8 * hl;
      const int col = col0 + 16 * t + lidx;
      float x = accs[t][r];
      if (bias) x += bias[col];
      if (EPI == EPI_SWISH) {
        x = x / (1.f + expf(-x));                    // x * sigmoid(x)
      } else if (EPI == EPI_RADIAL) {
        x *= rscale[row];                            // per-edge radial scale
      }
      D[d_base + (long long)row * d_rs + (long long)col * d_cs] = x;
    }
  }
}

__global__ void zero_f32(float* __restrict__ p, long long n) {
  long long i = (long long)blockIdx.x * blockDim.x + threadIdx.x;
  if (i < n) p[i] = 0.f;
}

// rscale[e] = mollifier_cutoff(r) / (r^5 + 1e-10)
__global__ void rscale_kernel(const float* __restrict__ rij,
                              float* __restrict__ rs, int E) {
  int e = blockIdx.x * blockDim.x + threadIdx.x;
  if (e >= E) return;
  float r = rij[e];
  float c = 0.f;
  if (r + 1e-7f < 5.0f) {
    float t = r * (1.0f / 5.0f);
    c = expf(1.0f - 1.0f / (1.0f - t * t));
  }
  float r2 = r * r;
  float r5 = r2 * r2 * r;
  rs[e] = c / (r5 + 1e-10f);
}

// Per-edge interaction: one block per edge, thread f handles feature f.
__global__ void edge_interact(const int* __restrict__ src,
                              const int* __restrict__ dst,
                              const float* __restrict__ mu,     // [N,F,3]
                              const float* __restrict__ rij,    // [E]
                              const float* __restrict__ vij,    // [E,3]
                              const float* __restrict__ radial, // [E,F] (pre-scaled)
                              const float* __restrict__ q,      // [N,F]
                              float* __restrict__ v_raw,        // [N,F]  accum
                              float* __restrict__ mvt_raw,      // [N,F,3] accum
                              int F) {
  const int e = blockIdx.x;
  const int f = threadIdx.x;
  const int s = src[e];
  const int d = dst[e];
  const float vx = vij[(long long)e * 3 + 0];
  const float vy = vij[(long long)e * 3 + 1];
  const float vz = vij[(long long)e * 3 + 2];
  const float r  = rij[e];
  const float r2 = r * r;

  const float* mus = mu + ((long long)s * F + f) * 3;
  const float* mud = mu + ((long long)d * F + f) * 3;
  const float msx = mus[0], msy = mus[1], msz = mus[2];
  const float mdx = mud[0], mdy = mud[1], mdz = mud[2];

  const float dot_s = msx * vx + msy * vy + msz * vz;
  const float dot_d = mdx * vx + mdy * vy + mdz * vz;
  const float rad = radial[(long long)e * F + f];
  const float qs  = q[(long long)s * F + f];
  const float qd  = q[(long long)d * F + f];

  const float tdx = rad * (r2 * mdx - 3.f * vx * dot_d);
  const float tdy = rad * (r2 * mdy - 3.f * vy * dot_d);
  const float tdz = rad * (r2 * mdz - 3.f * vz * dot_d);
  const float tsx = rad * (r2 * msx - 3.f * vx * dot_s);
  const float tsy = rad * (r2 * msy - 3.f * vy * dot_s);
  const float tsz = rad * (r2 * msz - 3.f * vz * dot_s);

  float* mvd = mvt_raw + ((long long)d * F + f) * 3;
  atomicAdd(&mvd[0], tdx * qs - tsx * qd);
  atomicAdd(&mvd[1], tdy * qs - tsy * qd);
  atomicAdd(&mvd[2], tdz * qs - tsz * qd);

  const float diag = (msx * mdx + msy * mdy + msz * mdz) * r2;
  const float v_edge = (diag - 3.f * dot_s * dot_d) * rad;
  atomicAdd(&v_raw[(long long)d * F + f], v_edge);
}

extern "C" void kernel_launch(void* const* d_in, const int* in_sizes, int n_in,
                              void* d_out, int out_size, void* d_ws, size_t ws_size,
                              hipStream_t stream) {
  // Inputs per setup_inputs order:
  const int*   src  = (const int*)d_in[0];
  const int*   dst  = (const int*)d_in[1];
  const float* mu   = (const float*)d_in[2];
  const float* rij  = (const float*)d_in[3];
  const float* vij  = (const float*)d_in[4];
  const float* f_ij = (const float*)d_in[5];
  const float* feat = (const float*)d_in[6];
  const float* We1  = (const float*)d_in[7];
  const float* be1  = (const float*)d_in[8];
  const float* We2  = (const float*)d_in[9];
  const float* be2  = (const float*)d_in[10];
  const float* Wq1  = (const float*)d_in[11];
  const float* bq1  = (const float*)d_in[12];
  const float* Wq2  = (const float*)d_in[13];
  const float* bq2  = (const float*)d_in[14];
  const float* Wv   = (const float*)d_in[15];
  const float* Wd1  = (const float*)d_in[16];
  const float* bd1  = (const float*)d_in[17];
  const float* Wd2  = (const float*)d_in[18];
  const float* bd2  = (const float*)d_in[19];

  const int E  = in_sizes[0];
  const int F  = in_sizes[8];            // be1 has F elements
  const int N  = in_sizes[6] / F;        // feat is [N,F]
  const int NB = in_sizes[5] / E;        // f_ij is [E,NB]

  float* ws = (float*)d_ws;
  const long long EF = (long long)E * F;
  const long long NF = (long long)N * F;
  // Workspace map (floats). buf0 [0,EF) holds h1_edge during the radial MLP,
  // then is reused for all node-sized accumulators (6*NF < EF here).
  float* h1_edge  = ws;                  // [E,F]
  float* radial   = ws + EF;             // [E,F]
  float* rscale   = ws + 2 * EF;         // [E]
  float* v_raw    = ws;                  // [N,F]      (reuses buf0)
  float* mvt_raw  = ws + NF;             // [N,F,3]
  float* q_buf    = ws + 4 * NF;         // [N,F]
  float* h_node   = ws + 5 * NF;         // [N,F]

  float* out_v   = (float*)d_out;        // [N,F]
  float* out_mvt = (float*)d_out;        // base; mvt starts at offset NF

  const int TPB = 256;                   // 8 waves/block
  const int WPB = TPB / 32;
  auto blocks_for = [&](int M, int Nc) {
    int ntiles = (M / 16) * (Nc / 64);
    return (ntiles + WPB - 1) / WPB;
  };

  // 1) h1_edge = swish(f_ij @ We1 + be1)            [E,NB]@[NB,F]
  wmma_gemm_f32<EPI_SWISH, true><<<blocks_for(E, F), TPB, 0, stream>>>(
      f_ij, 0, NB, 1, We1, be1, nullptr, h1_edge, 0, F, 1, E, NB, F);

  // 2) per-edge radial scale = cutoff(r)/(r^5+1e-10)
  rscale_kernel<<<(E + 255) / 256, 256, 0, stream>>>(rij, rscale, E);

  // 3) radial = (h1_edge @ We2 + be2) * rscale[e]   [E,F]@[F,F]
  wmma_gemm_f32<EPI_RADIAL, true><<<blocks_for(E, F), TPB, 0, stream>>>(
      h1_edge, 0, F, 1, We2, be2, rscale, radial, 0, F, 1, E, F, F);

  // 4) zero node accumulators (v_raw + mvt_raw = 4*NF floats, reuses buf0)
  zero_f32<<<(int)((4 * NF + 255) / 256), 256, 0, stream>>>(v_raw, 4 * NF);

  // 5) h_node = swish(feat @ Wq1 + bq1);  q = h_node @ Wq2 + bq2
  wmma_gemm_f32<EPI_SWISH, true><<<blocks_for(N, F), TPB, 0, stream>>>(
      feat, 0, F, 1, Wq1, bq1, nullptr, h_node, 0, F, 1, N, F, F);
  wmma_gemm_f32<EPI_NONE, true><<<blocks_for(N, F), TPB, 0, stream>>>(
      h_node, 0, F, 1, Wq2, bq2, nullptr, q_buf, 0, F, 1, N, F, F);

  // 6) edge gather/compute/scatter (atomics into v_raw, mvt_raw)
  edge_interact<<<E, F, 0, stream>>>(src, dst, mu, rij, vij, radial, q_buf,
                                     v_raw, mvt_raw, F);

  // 7) mvt_out[n,o,i] = sum_f mvt_raw[n,f,i] * Wv[f,o]   (3 strided GEMMs)
  for (int i = 0; i < 3; ++i) {
    wmma_gemm_f32<EPI_NONE, false><<<blocks_for(N, F), TPB, 0, stream>>>(
        mvt_raw, i, 3 * F, 3, Wv, nullptr, nullptr,
        out_mvt, NF + i, 3 * F, 3, N, F, F);
  }

  // 8) v = swish(v_raw @ Wd1 + bd1) @ Wd2 + bd2
  wmma_gemm_f32<EPI_SWISH, true><<<blocks_for(N, F), TPB, 0, stream>>>(
      v_raw, 0, F, 1, Wd1, bd1, nullptr, h_node, 0, F, 1, N, F, F);
  wmma_gemm_f32<EPI_NONE, true><<<blocks_for(N, F), TPB, 0, stream>>>(
      h_node, 0, F, 1, Wd2, bd2, nullptr, out_v, 0, F, 1, N, F, F);

  (void)n_in; (void)out_size; (void)ws_size;
}